// AdditiveFastFFlayer_32873679684248
// MI455X (gfx1250) — compile-verified
//
#include <hip/hip_runtime.h>
#include <hip/hip_bf16.h>

typedef __attribute__((ext_vector_type(2))) float v2f;
typedef __attribute__((ext_vector_type(8))) float v8f;
typedef __attribute__((ext_vector_type(4))) unsigned v4u;
typedef __attribute__((ext_vector_type(8))) int v8i_;
typedef __attribute__((ext_vector_type(4))) int v4i_;

#define NUM_HEAD 16
#define DIM_HEAD 64
#define IN_DIM   1024
#define SLEN     2048
#define BSZ      8
#define ROWS     (SLEN * BSZ)               /* 16384 */
#define QKV_N    (NUM_HEAD * 3 * DIM_HEAD)  /* 3072  */
#define CHUNK    16
#define NCHUNK   (SLEN / CHUNK)             /* 128   */
#define TSTRIDE  (BSZ * NUM_HEAD * 3 * DIM_HEAD) /* 24576 floats between consecutive t in qkv */

__device__ __forceinline__ v8f wmma4(v2f a, v2f b, v8f c) {
  // 8 args: (neg_a, A, neg_b, B, c_mod, C, reuse_a, reuse_b)
  return __builtin_amdgcn_wmma_f32_16x16x4_f32(false, a, false, b, (short)0, c,
                                               false, false);
}

// TDM: DMA one 64x64 f32 tile (row stride = rowStrideFloats) into LDS.
// D# group layouts per CDNA5 ISA ch.8 (CDNA5 ISA). 2D tensor, so
// trailing groups are zero. Tracked by TENSORcnt. 6-arg builtin (clang-23).
__device__ __forceinline__ void tdm_load_tile_64x64(const float* gptr,
                                                    unsigned lds_addr,
                                                    unsigned rowStrideFloats) {
  unsigned long long ga = (unsigned long long)(const void*)gptr;
  v4u g0;
  g0.x = 1u;                                    // count=1 (valid), user mode
  g0.y = lds_addr;                              // lds_addr [63:32]
  g0.z = (unsigned)ga;                          // global_addr [95:64]
  g0.w = (unsigned)((ga >> 32) & 0x01FFFFFFu)   // global_addr [120:96]
         | 0x80000000u;                         // type=2 ("image") [127:126]
  v8i_ g1;
  g1[0] = 0x00020000;                           // data_size=2 (4 bytes)
  g1[1] = 0;                                    // tensor_dim0[15:0] << 16
  g1[2] = 0x10;                                 // tensor_dim0[31:16]=0x10 (1M)
  g1[3] = (64 << 16) | 0x10;                    // tensor_dim1 hi=0x10, tile_dim0=64
  g1[4] = 64;                                   // tile_dim1=64, tile_dim2=0
  g1[5] = (int)rowStrideFloats;                 // tensor_dim0_stride[31:0]
  g1[6] = 0;                                    // stride hi / dim1_stride lo
  g1[7] = 0;
  v4i_ z4 = {0, 0, 0, 0};
  v8i_ z8 = {0, 0, 0, 0, 0, 0, 0, 0};
  __builtin_amdgcn_tensor_load_to_lds(g0, g1, z4, z4, z8, 0);
}

// ---------------------------------------------------------------------------
// 1) LayerNorm: h = (x - mu)/sqrt(var+eps)*gamma + beta ; one block per row
// ---------------------------------------------------------------------------
__global__ void ln_kernel(const float* __restrict__ x,
                          const float* __restrict__ gamma,
                          const float* __restrict__ beta,
                          float* __restrict__ h) {
  __shared__ float red[256];
  const int row = blockIdx.x;
  const int tid = threadIdx.x;
  const float* xr = x + (size_t)row * IN_DIM;
  float v[4];
  float s = 0.f;
#pragma unroll
  for (int i = 0; i < 4; ++i) { v[i] = xr[tid + 256 * i]; s += v[i]; }
  red[tid] = s;
  __syncthreads();
  for (int off = 128; off > 0; off >>= 1) {
    if (tid < off) red[tid] += red[tid + off];
    __syncthreads();
  }
  const float mu = red[0] * (1.f / IN_DIM);
  __syncthreads();
  float s2 = 0.f;
#pragma unroll
  for (int i = 0; i < 4; ++i) { float d = v[i] - mu; s2 += d * d; }
  red[tid] = s2;
  __syncthreads();
  for (int off = 128; off > 0; off >>= 1) {
    if (tid < off) red[tid] += red[tid + off];
    __syncthreads();
  }
  const float rstd = rsqrtf(red[0] * (1.f / IN_DIM) + 1e-5f);
  float* hr = h + (size_t)row * IN_DIM;
#pragma unroll
  for (int i = 0; i < 4; ++i) {
    int c = tid + 256 * i;
    hr[c] = (v[i] - mu) * rstd * gamma[c] + beta[c];
  }
}

// ---------------------------------------------------------------------------
// 2/6) C[M,N] = A[M,K] @ Bt[N,K]^T (+ optional residual).
//      256-thread block -> 128(M) x 64(N) tile. B panel (64x64) double-
//      buffered in LDS via the Tensor Data Mover (TENSORcnt pipelined).
//      Each wave: one 16-row M strip x 4 N-tiles (A operand reused 4x).
// ---------------------------------------------------------------------------
__global__ void __launch_bounds__(256)
gemm_nt_tiled(const float* __restrict__ A, const float* __restrict__ Bt,
              const float* __restrict__ resid, float* __restrict__ C,
              int N, int K, int use_resid) {
  __shared__ __align__(16) float Bp[2][64 * 64];  // [buf][n*64 + k] ; 32 KB
  const int tid = threadIdx.x;
  const int wv = tid >> 5, lane = tid & 31;
  const int r = lane & 15, hf = lane >> 4;
  const int nTilesN = N >> 6;
  const int bm = blockIdx.x / nTilesN;
  const int bn = blockIdx.x % nTilesN;
  const int m0 = bm * 128 + wv * 16;
  const int n0 = bn * 64;

  unsigned ldsbase[2];
  ldsbase[0] = (unsigned)(unsigned long long)(void*)&Bp[0][0];
  ldsbase[1] = (unsigned)(unsigned long long)(void*)&Bp[1][0];

  const float* arow = A + (size_t)(m0 + r) * K + 2 * hf;
  v8f acc[4];
#pragma unroll
  for (int nt = 0; nt < 4; ++nt) acc[nt] = (v8f){};

  // TDM issues are per-wave and ignore EXEC: only wave 0 drives the DMA.
  if (wv == 0) tdm_load_tile_64x64(Bt + (size_t)n0 * K, ldsbase[0], (unsigned)K);

  const int nKB = K >> 6;
  for (int kb = 0; kb < nKB; ++kb) {
    if (wv == 0) {
      if (kb + 1 < nKB) {
        tdm_load_tile_64x64(Bt + (size_t)n0 * K + ((kb + 1) << 6),
                            ldsbase[(kb + 1) & 1], (unsigned)K);
        __builtin_amdgcn_s_wait_tensorcnt(1);  // current panel landed
      } else {
        __builtin_amdgcn_s_wait_tensorcnt(0);
      }
    }
    __syncthreads();  // publish panel to all waves

    if (kb + 2 < nKB) __builtin_prefetch(arow + ((kb + 2) << 6), 0, 0);

    const float* bp = &Bp[kb & 1][0];
    const float* ak = arow + (kb << 6);
#pragma unroll
    for (int k = 0; k < 64; k += 4) {
      v2f a = *(const v2f*)(ak + k);
      v2f b0 = *(const v2f*)(bp + (0 * 16 + r) * 64 + k + 2 * hf);
      v2f b1 = *(const v2f*)(bp + (1 * 16 + r) * 64 + k + 2 * hf);
      v2f b2 = *(const v2f*)(bp + (2 * 16 + r) * 64 + k + 2 * hf);
      v2f b3 = *(const v2f*)(bp + (3 * 16 + r) * 64 + k + 2 * hf);
      acc[0] = wmma4(a, b0, acc[0]);
      acc[1] = wmma4(a, b1, acc[1]);
      acc[2] = wmma4(a, b2, acc[2]);
      acc[3] = wmma4(a, b3, acc[3]);
    }
    __syncthreads();  // done reading this buffer before it is re-staged
  }

#pragma unroll
  for (int nt = 0; nt < 4; ++nt) {
#pragma unroll
    for (int i = 0; i < 8; ++i) {
      size_t idx = (size_t)(m0 + i + 8 * hf) * N + n0 + nt * 16 + r;
      float v = acc[nt][i];
      if (use_resid) v += resid[idx];
      C[idx] = v;
    }
  }
}

// ---------------------------------------------------------------------------
// 3) q,k <- elu_p1_sum_norm (in place inside qkv buffer). One 64-thread block
//    per (s,b,h) row pair.
// ---------------------------------------------------------------------------
__global__ void qk_norm(float* __restrict__ qkv) {
  __shared__ float red[64];
  const int tid = threadIdx.x;
  float* base = qkv + (size_t)blockIdx.x * (3 * DIM_HEAD);
#pragma unroll
  for (int which = 0; which < 2; ++which) {
    float xv = base[which * DIM_HEAD + tid];
    float y = xv > 0.f ? xv + 1.f : __expf(xv);
    red[tid] = y;
    __syncthreads();
    if (tid < 32) red[tid] += red[tid + 32];
    __syncthreads();
    if (tid < 16) red[tid] += red[tid + 16];
    __syncthreads();
    if (tid < 8)  red[tid] += red[tid + 8];
    __syncthreads();
    if (tid < 4)  red[tid] += red[tid + 4];
    __syncthreads();
    if (tid < 2)  red[tid] += red[tid + 2];
    __syncthreads();
    if (tid < 1)  red[tid] += red[tid + 1];
    __syncthreads();
    const float s = red[0];
    __syncthreads();
    base[which * DIM_HEAD + tid] = y / (s + 1e-5f);
  }
}

// ---------------------------------------------------------------------------
// 4) Inter-chunk recurrence. One block (4 waves) per (b,h) chain.
//    W^T kept in LDS across all 128 chunks; the chunk's 16x192 q/k/v slice is
//    double-buffered into LDS one chunk ahead with async-to-LDS copies so the
//    serial scan never stalls on strided global loads. Per chunk:
//      O_inter = Q_c @ W      (each wave one 16x16 e-slab, 16 WMMA)
//      W      += K_c^T @ V_c  (each wave a 16-row d-slab, 16 WMMA)
// ---------------------------------------------------------------------------
__global__ void recur_inter(const float* __restrict__ qkv,
                            const float* __restrict__ state,
                            float* __restrict__ o,
                            float* __restrict__ Wfin) {
  __shared__ float WT[DIM_HEAD * DIM_HEAD];          // WT[e*64 + d] = W[d][e]
  __shared__ __align__(16) float CQ[2][CHUNK * 192]; // chunk q/k/v slice, 24 KB
  const int bh = blockIdx.x;                         // b*16 + h
  const int b = bh >> 4, h = bh & 15;
  const int tid  = threadIdx.x;
  const int wv   = tid >> 5;
  const int lane = tid & 31;
  const int r  = lane & 15;
  const int hf = lane >> 4;

  unsigned cqbase[2];
  cqbase[0] = (unsigned)(unsigned long long)(void*)&CQ[0][0];
  cqbase[1] = (unsigned)(unsigned long long)(void*)&CQ[1][0];

  for (int i = tid; i < DIM_HEAD * DIM_HEAD; i += 128) {
    int d = i >> 6, e = i & 63;
    WT[e * 64 + d] = state[(size_t)bh * (DIM_HEAD * DIM_HEAD) + i];
  }

  const float* cb = qkv + (size_t)bh * (3 * DIM_HEAD);

  // Async-stage rows t0..t0+15 (16 x 192 floats) of this chain into CQ[buf].
  auto stage_chunk = [&](int buf, int t0) {
#pragma unroll
    for (int j = 0; j < 6; ++j) {
      int cc = tid + 128 * j;        // 768 b128 chunks, 48 per row
      int t = cc / 48;
      int col = (cc % 48) * 4;       // float index within row
      unsigned lds  = cqbase[buf] + (unsigned)(((t * 192) + col) << 2);
      unsigned goff = (unsigned)((((t0 + t) * TSTRIDE) + col) << 2);
      asm volatile("global_load_async_to_lds_b128 %0, %1, %2"
                   :
                   : "v"(lds), "v"(goff), "s"(cb)
                   : "memory");
    }
  };

  stage_chunk(0, 0);
  for (int c = 0; c < NCHUNK; ++c) {
    const int t0 = c * CHUNK;
    if (c + 1 < NCHUNK) {
      stage_chunk((c + 1) & 1, t0 + CHUNK);
      asm volatile("s_wait_asynccnt 6" ::: "memory");  // current chunk landed
    } else {
      asm volatile("s_wait_asynccnt 0" ::: "memory");
    }
    __syncthreads();
    const float* cq = &CQ[c & 1][0];

    // ---- O_inter tile: rows t0..t0+15, cols e0..e0+15 ----
    const int e0 = wv * 16;
    const float* qb = cq + r * 192 + 2 * hf;             // q comp
    const float* wb = &WT[(e0 + r) * 64 + 2 * hf];
    v8f acc = {};
#pragma unroll
    for (int k = 0; k < DIM_HEAD; k += 4)
      acc = wmma4(*(const v2f*)(qb + k), *(const v2f*)(wb + k), acc);
#pragma unroll
    for (int i = 0; i < 8; ++i) {
      int t = t0 + i + 8 * hf;
      o[(size_t)t * (BSZ * IN_DIM) + b * IN_DIM + h * DIM_HEAD + e0 + r] = acc[i];
    }
    __syncthreads();  // all W reads done before updates

    // ---- W += K^T V : wave owns d-slab d0..d0+15, loops 4 e-tiles ----
    const int d0 = wv * 16;
    const float* kbp = cq + (2 * hf) * 192 + DIM_HEAD + d0 + r;
    v2f ka[4];
#pragma unroll
    for (int ks = 0; ks < 4; ++ks) {
      ka[ks].x = kbp[(4 * ks) * 192];
      ka[ks].y = kbp[(4 * ks) * 192 + 192];
    }
#pragma unroll
    for (int et = 0; et < 4; ++et) {
      const int ee0 = et * 16;
      const float* vbp = cq + (2 * hf) * 192 + 2 * DIM_HEAD + ee0 + r;
      v8f s = {};
#pragma unroll
      for (int ks = 0; ks < 4; ++ks) {
        v2f bb;
        bb.x = vbp[(4 * ks) * 192];
        bb.y = vbp[(4 * ks) * 192 + 192];
        s = wmma4(ka[ks], bb, s);
      }
#pragma unroll
      for (int i = 0; i < 8; ++i) {  // disjoint addresses across lanes/waves
        int d = d0 + i + 8 * hf;
        int e = ee0 + r;
        WT[e * 64 + d] += s[i];
      }
    }
    __syncthreads();  // updates visible / CQ[c&1] free before next stage
  }

  for (int i = tid; i < DIM_HEAD * DIM_HEAD; i += 128) {
    int d = i >> 6, e = i & 63;
    Wfin[(size_t)bh * (DIM_HEAD * DIM_HEAD) + i] = WT[e * 64 + d];
  }
}

// ---------------------------------------------------------------------------
// 5) Intra-chunk causal attention: O_c += mask(Q_c K_c^T) V_c.
//    One wave per (chain, chunk); P staged via LDS to re-layout C->A operand.
// ---------------------------------------------------------------------------
__global__ void intra_chunk(const float* __restrict__ qkv, float* __restrict__ o) {
  __shared__ float Pl[4 * 256];
  const int tid  = threadIdx.x;
  const int wv   = tid >> 5;
  const int lane = tid & 31;
  const int r  = lane & 15;
  const int hf = lane >> 4;
  const int job = blockIdx.x * 4 + wv;
  const int chain = job >> 7;  // b*16 + h
  const int c = job & (NCHUNK - 1);
  const int b = chain >> 4, h = chain & 15;
  const int t0 = c * CHUNK;
  const float* cb = qkv + (size_t)chain * (3 * DIM_HEAD);

  // P = Q K^T over d (K-dim = 64)
  const float* qrow = cb + (size_t)(t0 + r) * TSTRIDE + 2 * hf;
  const float* krow = cb + (size_t)(t0 + r) * TSTRIDE + DIM_HEAD + 2 * hf;
  v8f p = {};
#pragma unroll
  for (int k = 0; k < DIM_HEAD; k += 4)
    p = wmma4(*(const v2f*)(qrow + k), *(const v2f*)(krow + k), p);

  // causal mask (keep s <= t) and stage to LDS in row-major [t][s]
  float* pl = &Pl[wv * 256];
#pragma unroll
  for (int i = 0; i < 8; ++i) {
    int t = i + 8 * hf;
    pl[t * 16 + r] = (r <= t) ? p[i] : 0.f;
  }
  __syncthreads();

  // A-operands of P (K-dim = s, 16)
  v2f pa[4];
#pragma unroll
  for (int ks = 0; ks < 4; ++ks)
    pa[ks] = *(const v2f*)(pl + r * 16 + 4 * ks + 2 * hf);

#pragma unroll
  for (int et = 0; et < 4; ++et) {
    const int e0 = et * 16;
    const float* vb =
        cb + (size_t)(t0 + 2 * hf) * TSTRIDE + 2 * DIM_HEAD + e0 + r;
    v8f acc;
#pragma unroll
    for (int i = 0; i < 8; ++i) {  // C init = O_inter from recur_inter
      int t = t0 + i + 8 * hf;
      acc[i] = o[(size_t)t * (BSZ * IN_DIM) + b * IN_DIM + h * DIM_HEAD + e0 + r];
    }
#pragma unroll
    for (int ks = 0; ks < 4; ++ks) {
      v2f bb;
      bb.x = vb[(size_t)(4 * ks) * TSTRIDE];
      bb.y = vb[(size_t)(4 * ks + 1) * TSTRIDE];
      acc = wmma4(pa[ks], bb, acc);
    }
#pragma unroll
    for (int i = 0; i < 8; ++i) {
      int t = t0 + i + 8 * hf;
      o[(size_t)t * (BSZ * IN_DIM) + b * IN_DIM + h * DIM_HEAD + e0 + r] = acc[i];
    }
  }
}

// ---------------------------------------------------------------------------
extern "C" void kernel_launch(void* const* d_in, const int* in_sizes, int n_in,
                              void* d_out, int out_size, void* d_ws, size_t ws_size,
                              hipStream_t stream) {
  const float* x     = (const float*)d_in[0];
  const float* state = (const float*)d_in[1];
  const float* W_qkv = (const float*)d_in[2];
  const float* W_out = (const float*)d_in[3];
  const float* gamma = (const float*)d_in[4];
  const float* beta  = (const float*)d_in[5];

  float* out  = (float*)d_out;                                  // [S,B,1024]
  float* Wfin = out + (size_t)SLEN * BSZ * IN_DIM;              // [B,H,64,64]

  float* h   = (float*)d_ws;                                    //  64 MB
  float* qkv = h + (size_t)ROWS * IN_DIM;                       // 192 MB
  float* o   = qkv + (size_t)ROWS * QKV_N;                      //  64 MB

  // 1) h = LayerNorm(x)
  ln_kernel<<<ROWS, 256, 0, stream>>>(x, gamma, beta, h);

  // 2) qkv = h @ W_qkv^T   (LDS-tiled WMMA + TDM B-panel staging)
  {
    int blocks = (ROWS / 128) * (QKV_N / 64);
    gemm_nt_tiled<<<blocks, 256, 0, stream>>>(h, W_qkv, nullptr, qkv, QKV_N,
                                              IN_DIM, 0);
  }

  // 3) q,k normalize in place
  qk_norm<<<ROWS * NUM_HEAD, 64, 0, stream>>>(qkv);

  // 4) inter-chunk recurrence + final fast-weight state
  recur_inter<<<BSZ * NUM_HEAD, 128, 0, stream>>>(qkv, state, o, Wfin);

  // 5) intra-chunk causal attention (fully parallel)
  intra_chunk<<<(BSZ * NUM_HEAD * NCHUNK) / 4, 128, 0, stream>>>(qkv, o);

  // 6) out = o @ W_out^T + x
  {
    int blocks = (ROWS / 128) * (IN_DIM / 64);
    gemm_nt_tiled<<<blocks, 256, 0, stream>>>(o, W_out, x, out, IN_DIM,
                                              IN_DIM, 1);
  }
}